// GCNmfConv_56255481643194
// MI455X (gfx1250) — compile-verified
//
#include <hip/hip_runtime.h>
#include <math.h>

typedef __attribute__((ext_vector_type(2))) float v2f;
typedef __attribute__((ext_vector_type(8))) float v8f;

#define IN_DIM 256
#define OUT_DIM 128
#define KCOMP 5

// ---------------------------------------------------------------------------
// init: cnt[n]=1 (self loop), cursor[n]=0
__global__ void init_kernel(int* cnt, int* cursor, int n) {
    int i = blockIdx.x * blockDim.x + threadIdx.x;
    if (i < n) { cnt[i] = 1; cursor[i] = 0; }
}

// count in-degree per destination row (row = edge_index[0])
__global__ void count_kernel(const int* __restrict__ rows, int* cnt, int E) {
    int e = blockIdx.x * blockDim.x + threadIdx.x;
    if (e < E) atomicAdd(&cnt[rows[e]], 1);
}

// dinv[n] = 1/sqrt(deg[n])  (deg >= 1 always due to self loop)
__global__ void dinv_kernel(const int* __restrict__ cnt, float* dinv, int n) {
    int i = blockIdx.x * blockDim.x + threadIdx.x;
    if (i < n) {
        float d = (float)cnt[i];
        dinv[i] = rsqrtf(d);
    }
}

// single-block exclusive scan: offsets[0]=0; offsets[i+1]=sum cnt[0..i]
__global__ void scan_kernel(const int* __restrict__ cnt, int* offsets, int n) {
    __shared__ int sh[1024];
    __shared__ int carry;
    int t = threadIdx.x;
    if (t == 0) carry = 0;
    __syncthreads();
    for (int base = 0; base < n; base += 1024) {
        int i = base + t;
        int v = (i < n) ? cnt[i] : 0;
        sh[t] = v;
        __syncthreads();
        for (int off = 1; off < 1024; off <<= 1) {
            int add = (t >= off) ? sh[t - off] : 0;
            __syncthreads();
            sh[t] += add;
            __syncthreads();
        }
        if (i < n) offsets[i + 1] = carry + sh[t];
        __syncthreads();
        if (t == 1023) carry += sh[1023];
        __syncthreads();
    }
    if (t == 0) offsets[0] = 0;
}

// scatter edges (+ self loops) into CSR with normalized weights
__global__ void scatter_kernel(const int* __restrict__ rows, const int* __restrict__ cols,
                               const float* __restrict__ dinv, const int* __restrict__ offsets,
                               int* cursor, int* ccol, float* cw, int E, int n) {
    int e = blockIdx.x * blockDim.x + threadIdx.x;
    if (e < E) {
        int r = rows[e], c = cols[e];
        int p = offsets[r] + atomicAdd(&cursor[r], 1);
        ccol[p] = c;
        cw[p] = dinv[r] * dinv[c];
    } else if (e < E + n) {
        int r = e - E;
        int p = offsets[r] + atomicAdd(&cursor[r], 1);
        ccol[p] = r;
        cw[p] = dinv[r] * dinv[r];
    }
}

// ---------------------------------------------------------------------------
// GMM responsibilities: one wave32 per node. Constant terms cancel in softmax.
__global__ void gamma_kernel(const float* __restrict__ x, const float* __restrict__ logp,
                             const float* __restrict__ means, const float* __restrict__ logvars,
                             float* __restrict__ gamma, int n) {
    int wid  = (blockIdx.x * blockDim.x + threadIdx.x) >> 5;
    int lane = threadIdx.x & 31;
    if (wid >= n) return;
    const float* xr = x + (size_t)wid * IN_DIM;
    float acc[KCOMP] = {0.f, 0.f, 0.f, 0.f, 0.f};
    for (int i = lane; i < IN_DIM; i += 32) {
        float xv = xr[i];
        bool ok = (xv == xv);  // not NaN
#pragma unroll
        for (int k = 0; k < KCOMP; k++) {
            float mk = means[k * IN_DIM + i];
            float iv = expf(-logvars[k * IN_DIM + i]);
            float d = xv - mk;
            float contrib = d * d * iv;
            acc[k] += ok ? contrib : 0.f;
        }
    }
#pragma unroll
    for (int off = 16; off > 0; off >>= 1)
#pragma unroll
        for (int k = 0; k < KCOMP; k++) acc[k] += __shfl_down(acc[k], off, 32);
    if (lane == 0) {
        float l[KCOMP], m = -1e30f;
#pragma unroll
        for (int k = 0; k < KCOMP; k++) { l[k] = logp[k] - 0.5f * acc[k]; m = fmaxf(m, l[k]); }
        float s = 0.f;
#pragma unroll
        for (int k = 0; k < KCOMP; k++) { l[k] = expf(l[k] - m); s += l[k]; }
        float inv = 1.f / s;
#pragma unroll
        for (int k = 0; k < KCOMP; k++) gamma[(size_t)k * n + wid] = l[k] * inv;
    }
}

// ---------------------------------------------------------------------------
// WMMA GEMM: per block one 16-row tile, 8 waves = 8 column groups of 16.
// For each K-step of 4: 5 components x 2 matrices = 10 v_wmma_f32_16x16x4_f32.
// f32 A/B frag layout (ISA 7.12.2): VGPR v, lane-half h -> K = 2h+v, M/N = lane&15.
// C/D: VGPR v, half h -> M = v+8h, N = lane&15.
// All LDS reads are hoisted unconditionally so NaN-imputation lowers to
// v_cndmask (branch-free inner loop; WMMA needs EXEC all-ones anyway).
__global__ __launch_bounds__(256) void gemm_wmma_kernel(
    const float* __restrict__ x, const float* __restrict__ means,
    const float* __restrict__ logvars, const float* __restrict__ weight,
    const float* __restrict__ bias, float* __restrict__ tx, float* __restrict__ tcov, int n) {
    __shared__ float sx[16 * 257];          // padded to break bank conflicts
    __shared__ float sm[KCOMP * IN_DIM];
    __shared__ float sv[KCOMP * IN_DIM];

    int rowBase = blockIdx.x * 16;
    int tid = threadIdx.x;

    for (int i = tid; i < 16 * IN_DIM; i += 256) {
        int r = i >> 8, c = i & 255;
        int rr = rowBase + r; rr = rr < n ? rr : n - 1;
        sx[r * 257 + c] = x[(size_t)rr * IN_DIM + c];
    }
    for (int i = tid; i < KCOMP * IN_DIM; i += 256) {
        sm[i] = means[i];
        sv[i] = expf(logvars[i]);            // variances
    }
    __syncthreads();

    int w    = tid >> 5;   // wave -> column group
    int lane = tid & 31;
    int h    = lane >> 4;  // lane half
    int ln   = lane & 15;
    int col  = w * 16 + ln;

    const float* sxr  = sx + ln * 257;
    const float* wcol = weight + col;

    v8f zero = {0.f, 0.f, 0.f, 0.f, 0.f, 0.f, 0.f, 0.f};
    v8f accX[KCOMP], accV[KCOMP];
#pragma unroll
    for (int c = 0; c < KCOMP; c++) { accX[c] = zero; accV[c] = zero; }

    for (int kk = 0; kk < IN_DIM; kk += 4) {
        int kA = kk + 2 * h;                 // this lane's two K indices: kA, kA+1
        float x0 = sxr[kA];
        float x1 = sxr[kA + 1];
        bool nan0 = !(x0 == x0);
        bool nan1 = !(x1 == x1);
        float bx = wcol[(size_t)kA * OUT_DIM];
        float by = wcol[(size_t)(kA + 1) * OUT_DIM];
        v2f b   = {bx, by};
        v2f bsq = {bx * bx, by * by};
#pragma unroll
        for (int c = 0; c < KCOMP; c++) {
            // unconditional broadcast LDS loads -> selects become v_cndmask
            float m0  = sm[c * IN_DIM + kA];
            float m1  = sm[c * IN_DIM + kA + 1];
            float vv0 = sv[c * IN_DIM + kA];
            float vv1 = sv[c * IN_DIM + kA + 1];
            v2f aX = {nan0 ? m0 : x0, nan1 ? m1 : x1};
            v2f aV = {nan0 ? vv0 : 0.f, nan1 ? vv1 : 0.f};
            accX[c] = __builtin_amdgcn_wmma_f32_16x16x4_f32(
                false, aX, false, b,   (short)0, accX[c], false, false);
            accV[c] = __builtin_amdgcn_wmma_f32_16x16x4_f32(
                false, aV, false, bsq, (short)0, accV[c], false, false);
        }
    }

    float bv = bias[col];
#pragma unroll
    for (int c = 0; c < KCOMP; c++) {
#pragma unroll
        for (int v = 0; v < 8; v++) {
            int row = rowBase + v + 8 * h;
            if (row < n) {
                tx  [((size_t)c * n + row) * OUT_DIM + col] = accX[c][v] + bv;
                tcov[((size_t)c * n + row) * OUT_DIM + col] = accV[c][v];
            }
        }
    }
}

// ---------------------------------------------------------------------------
__device__ __forceinline__ float ex_relu(float mu, float sigma) {
    float sg = (sigma == 0.f) ? 1e-10f : sigma;
    float ss = sqrtf(sg);
    float w  = mu / ss;
    float nr = ss * (expf(-0.5f * w * w) * 0.39894228040143267f
                     + 0.5f * w * (1.f + erff(w * 0.70710678118654752f)));
    return (sigma == 0.f) ? fmaxf(mu, 0.f) : nr;
}

// Fused CSR-gather SpMM + ex_relu + gamma mixing. One block per node,
// thread = output dim. t_x/t_cov (102MB) fit in the 192MB L2 -> gathers hit L2.
__global__ __launch_bounds__(128) void spmm_mix_kernel(
    const float* __restrict__ tx, const float* __restrict__ tcov,
    const int* __restrict__ offsets, const int* __restrict__ ccol,
    const float* __restrict__ cw, const float* __restrict__ gamma,
    float* __restrict__ out, int n) {
    int node = blockIdx.x;
    int o = threadIdx.x;
    int s = offsets[node], e = offsets[node + 1];
    float ax[KCOMP] = {0.f, 0.f, 0.f, 0.f, 0.f};
    float av[KCOMP] = {0.f, 0.f, 0.f, 0.f, 0.f};
    size_t plane = (size_t)n * OUT_DIM;
    for (int p = s; p < e; p++) {
        int c = ccol[p];
        float wv = cw[p];
        float w2 = wv * wv;
        size_t base = (size_t)c * OUT_DIM + o;
#pragma unroll
        for (int k = 0; k < KCOMP; k++) {
            ax[k] = fmaf(wv, tx[(size_t)k * plane + base], ax[k]);
            av[k] = fmaf(w2, tcov[(size_t)k * plane + base], av[k]);
        }
    }
    float res = 0.f;
#pragma unroll
    for (int k = 0; k < KCOMP; k++) {
        float g = gamma[(size_t)k * n + node];
        res += g * ex_relu(ax[k], av[k]);
    }
    out[(size_t)node * OUT_DIM + o] = res;
}

// ---------------------------------------------------------------------------
extern "C" void kernel_launch(void* const* d_in, const int* in_sizes, int n_in,
                              void* d_out, int out_size, void* d_ws, size_t ws_size,
                              hipStream_t stream) {
    const float* x       = (const float*)d_in[0];
    const int*   ei      = (const int*)d_in[1];   // [2, E]: rows then cols
    const float* logp    = (const float*)d_in[2];
    const float* means   = (const float*)d_in[3];
    const float* logvars = (const float*)d_in[4];
    const float* weight  = (const float*)d_in[5];
    const float* bias    = (const float*)d_in[6];
    float* out = (float*)d_out;

    const int N = in_sizes[0] / IN_DIM;
    const int E = in_sizes[1] / 2;
    const int NNZ = E + N;

    // carve workspace (256B aligned)
    char* p = (char*)d_ws;
    auto carve = [&](size_t bytes) -> void* {
        void* r = (void*)p;
        p += (bytes + 255) & ~(size_t)255;
        return r;
    };
    float* tx      = (float*)carve(sizeof(float) * (size_t)KCOMP * N * OUT_DIM);
    float* tcov    = (float*)carve(sizeof(float) * (size_t)KCOMP * N * OUT_DIM);
    float* gamma   = (float*)carve(sizeof(float) * (size_t)KCOMP * N);
    float* dinv    = (float*)carve(sizeof(float) * N);
    int*   cnt     = (int*)  carve(sizeof(int) * N);
    int*   offsets = (int*)  carve(sizeof(int) * (N + 1));
    int*   cursor  = (int*)  carve(sizeof(int) * N);
    int*   ccol    = (int*)  carve(sizeof(int) * NNZ);
    float* cw      = (float*)carve(sizeof(float) * NNZ);

    const int* rows = ei;
    const int* cols = ei + E;

    init_kernel<<<(N + 255) / 256, 256, 0, stream>>>(cnt, cursor, N);
    count_kernel<<<(E + 255) / 256, 256, 0, stream>>>(rows, cnt, E);
    dinv_kernel<<<(N + 255) / 256, 256, 0, stream>>>(cnt, dinv, N);
    scan_kernel<<<1, 1024, 0, stream>>>(cnt, offsets, N);
    scatter_kernel<<<(NNZ + 255) / 256, 256, 0, stream>>>(rows, cols, dinv, offsets,
                                                          cursor, ccol, cw, E, N);
    gamma_kernel<<<(N + 7) / 8, 256, 0, stream>>>(x, logp, means, logvars, gamma, N);
    gemm_wmma_kernel<<<(N + 15) / 16, 256, 0, stream>>>(x, means, logvars, weight, bias,
                                                        tx, tcov, N);
    spmm_mix_kernel<<<N, 128, 0, stream>>>(tx, tcov, offsets, ccol, cw, gamma, out, N);
}